// XCA_86689619902842
// MI455X (gfx1250) — compile-verified
//
#include <hip/hip_runtime.h>
#include <hip/hip_bf16.h>

// XCA (cross-covariance attention) for MI455X / gfx1250.
// fp32 V_WMMA_F32_16X16X4_F32 pipeline (memory-bound: ~1.5 GB HBM traffic
// @ 23.3 TB/s ~= 65us; 245 GFLOP is negligible at matrix-core rates).
// Round-2 change: big GEMMs use the CDNA5 async-to-LDS path
// (global_load_async_to_lds_b128 + s_wait_asynccnt, double-buffered LDS
// 128x128 tiles) so operand traffic crosses L2->WGP once per block, and
// WMMA reads come from LDS instead of serialized global loads.
//
// Workspace (floats): q | k | v  ([B*H*48, 3136] each) | attn [B*H,48,48]
// q region is reused as O ([B,N,C]) after kernel 2.  Total ~465 MB.

typedef __attribute__((ext_vector_type(2))) float v2f;
typedef __attribute__((ext_vector_type(4))) float v4f;
typedef __attribute__((ext_vector_type(8))) float v8f;

#define B_   16
#define N_   3136
#define C_   768
#define H_   16
#define DH_  48
#define M_   (B_ * N_)    // 50176
#define C3_  (3 * C_)     // 2304

#define KC_   16          // k-panel depth staged per LDS buffer
#define LDA_  20          // padded LDS row stride (floats): bank-conflict-free
                          // for ds_load_b64 fragments, 80B rows keep b128 align
#define LDATT_ 52         // padded stride for 48-wide attn tile

static __device__ __forceinline__ v8f wmma_f32(v2f a, v2f b, v8f c) {
  // D(16x16,f32) += A(16x4,f32) * B(4x16,f32)
  return __builtin_amdgcn_wmma_f32_16x16x4_f32(
      false, a, false, b, (short)0, c, false, false);
}

// CDNA5 async copy: 16B per lane, global -> LDS, tracked by ASYNCcnt.
static __device__ __forceinline__ void async_b128(void* lds, const void* g) {
  asm volatile("global_load_async_to_lds_b128 %0, %1, off"
               :: "v"((unsigned)(unsigned long long)lds),
                  "v"((unsigned long long)g)
               : "memory");
}
#define WAIT_ASYNC_4() asm volatile("s_wait_asynccnt 0x4" ::: "memory")
#define WAIT_ASYNC_0() asm volatile("s_wait_asynccnt 0x0" ::: "memory")

// -------------------------------------------------------------------------
// Kernel 1: qkv = x @ w_qkv^T, stored transposed into q/k/v [B*H*dh, N].
// 256 threads (8 waves), 128x128 block tile, K panels of 16 double-buffered
// in LDS via async-to-LDS.  Wave w owns rows [w*16, w*16+16) x 128 cols.
// -------------------------------------------------------------------------
__global__ __launch_bounds__(256) void k1_qkv(const float* __restrict__ X,
                                              const float* __restrict__ W,
                                              float* __restrict__ Q,
                                              float* __restrict__ K,
                                              float* __restrict__ V) {
  __shared__ float sA[2][128 * LDA_];
  __shared__ float sB[2][128 * LDA_];

  const int tid  = threadIdx.x;
  const int lane = tid & 31;
  const int wave = tid >> 5;
  const int lr   = lane & 15;
  const int lh   = lane >> 4;

  const int mBase = blockIdx.y * 128;
  const int jBase = blockIdx.x * 128;

  // async segment assignment: thread -> (row, 16B segment), 4 loads/chunk
  const int r0 = tid >> 2;            // 0..63
  const int s0 = (tid & 3) * 4;       // 0,4,8,12 floats
  const float* gA0 = X + (mBase + r0)      * C_ + s0;
  const float* gA1 = X + (mBase + r0 + 64) * C_ + s0;
  const float* gB0 = W + (jBase + r0)      * C_ + s0;
  const float* gB1 = W + (jBase + r0 + 64) * C_ + s0;

  v8f acc[8] = {};

  // prefetch first panel
  async_b128(&sA[0][r0 * LDA_ + s0],        gA0);
  async_b128(&sA[0][(r0 + 64) * LDA_ + s0], gA1);
  async_b128(&sB[0][r0 * LDA_ + s0],        gB0);
  async_b128(&sB[0][(r0 + 64) * LDA_ + s0], gB1);

  const int NK = C_ / KC_;  // 48
  for (int i = 0; i < NK; ++i) {
    if (i + 1 < NK) {
      const int kc = (i + 1) * KC_;
      const int nb = (i + 1) & 1;
      async_b128(&sA[nb][r0 * LDA_ + s0],        gA0 + kc);
      async_b128(&sA[nb][(r0 + 64) * LDA_ + s0], gA1 + kc);
      async_b128(&sB[nb][r0 * LDA_ + s0],        gB0 + kc);
      async_b128(&sB[nb][(r0 + 64) * LDA_ + s0], gB1 + kc);
      WAIT_ASYNC_4();  // panel i landed (async loads complete in order)
    } else {
      WAIT_ASYNC_0();
    }
    __syncthreads();

    const float* As = &sA[i & 1][(wave * 16 + lr) * LDA_ + 2 * lh];
    const float* Bs = &sB[i & 1][lr * LDA_ + 2 * lh];
#pragma unroll
    for (int kk = 0; kk < KC_; kk += 4) {
      v2f a = *(const v2f*)(As + kk);
#pragma unroll
      for (int nt = 0; nt < 8; ++nt) {
        v2f b = *(const v2f*)(Bs + nt * 16 * LDA_ + kk);
        acc[nt] = wmma_f32(a, b, acc[nt]);
      }
    }
    __syncthreads();
  }

  // Transposed store into [B*H*dh, N]; D rows map to consecutive n.
  const int m0w = mBase + wave * 16;     // 16-aligned, 3136%16==0 => one b
  const int b   = m0w / N_;
  const int n   = (m0w % N_) + lh * 8;
#pragma unroll
  for (int nt = 0; nt < 8; ++nt) {
    const int j  = jBase + nt * 16 + lr;  // 16 | 48: tile never splits h
    const int s  = j / C_;
    const int jr = j % C_;
    const int h  = jr / DH_;
    const int d  = jr % DH_;
    float* base = (s == 0 ? Q : (s == 1 ? K : V)) +
                  ((b * H_ + h) * DH_ + d) * N_ + n;
    v4f lo = {acc[nt][0], acc[nt][1], acc[nt][2], acc[nt][3]};
    v4f hi = {acc[nt][4], acc[nt][5], acc[nt][6], acc[nt][7]};
    *(v4f*)(base)     = lo;
    *(v4f*)(base + 4) = hi;
  }
}

// -------------------------------------------------------------------------
// Kernel 2: per (b,h): row L2 norms of q,k; S = q @ k^T via WMMA (3x3 tiles,
// one per wave, 9 waves); scale by rinv_q*rinv_k*temperature; softmax rows.
// -------------------------------------------------------------------------
__global__ __launch_bounds__(288) void k2_attn(const float* __restrict__ Q,
                                               const float* __restrict__ Km,
                                               const float* __restrict__ temp,
                                               float* __restrict__ attn) {
  __shared__ float rsum[96];
  __shared__ float rinv[96];
  __shared__ float sS[DH_ * 49];

  const int tid = threadIdx.x;
  const int bh  = blockIdx.x;
  const int h   = bh & 15;
  const float* qb = Q  + (size_t)bh * DH_ * N_;
  const float* kb = Km + (size_t)bh * DH_ * N_;

  if (tid < 96) rsum[tid] = 0.0f;
  __syncthreads();

  {  // row-norm partials: 96 rows x 3 chunks
    const int row   = tid % 96;
    const int chunk = tid / 96;
    const float* src = (row < DH_) ? (qb + row * N_) : (kb + (row - DH_) * N_);
    const int nb = chunk * 1046;
    const int ne = (nb + 1046 < N_) ? nb + 1046 : N_;
    float s = 0.0f;
    for (int n = nb; n < ne; ++n) { float t = src[n]; s += t * t; }
    atomicAdd(&rsum[row], s);
  }
  __syncthreads();
  if (tid < 96) rinv[tid] = 1.0f / fmaxf(sqrtf(rsum[tid]), 1e-12f);
  __syncthreads();

  const int lane = tid & 31;
  const int wave = tid >> 5;
  const int lr   = lane & 15;
  const int lh   = lane >> 4;
  const int d0   = (wave / 3) * 16;
  const int e0   = (wave % 3) * 16;

  const float* aptr = qb + (d0 + lr) * N_ + 2 * lh;
  const float* bptr = kb + (e0 + lr) * N_ + 2 * lh;
  v8f c = {};
#pragma unroll 4
  for (int n0 = 0; n0 < N_; n0 += 4) {
    v2f a = *(const v2f*)(aptr + n0);
    v2f b = *(const v2f*)(bptr + n0);
    c = wmma_f32(a, b, c);
  }

  const float tm = temp[h];
#pragma unroll
  for (int rr = 0; rr < 8; ++rr) {
    const int d = d0 + rr + 8 * lh;
    const int e = e0 + lr;
    sS[d * 49 + e] = c[rr] * rinv[d] * rinv[48 + e] * tm;
  }
  __syncthreads();

  if (tid < DH_) {
    float mx = -1e30f;
    for (int e = 0; e < DH_; ++e) mx = fmaxf(mx, sS[tid * 49 + e]);
    float sum = 0.0f;
    for (int e = 0; e < DH_; ++e) sum += __expf(sS[tid * 49 + e] - mx);
    const float inv = 1.0f / sum;
    float* dst = attn + (size_t)bh * DH_ * DH_ + tid * DH_;
    for (int e = 0; e < DH_; ++e) dst[e] = __expf(sS[tid * 49 + e] - mx) * inv;
  }
}

// -------------------------------------------------------------------------
// Kernel 3: O[b,n,h*48+d] = sum_e attn[b,h,d,e] * v[b,h,e,n].
// One wave: attn (48x48) and V tile (48x64) staged to LDS with async-to-LDS,
// then 12 k-steps of pure ds_load -> WMMA.  D rows map to consecutive d =>
// b128 stores straight into [B,N,C].
// -------------------------------------------------------------------------
__global__ __launch_bounds__(32) void k3_av(const float* __restrict__ attn,
                                            const float* __restrict__ V,
                                            float* __restrict__ O) {
  __shared__ float sAttn[DH_ * LDATT_];
  __shared__ float sV[DH_ * 64];

  const int tid = threadIdx.x;
  const int bh  = blockIdx.y;
  const int b   = bh >> 4;
  const int h   = bh & 15;
  const int n0  = blockIdx.x * 64;

  const float* ab = attn + (size_t)bh * DH_ * DH_;
#pragma unroll
  for (int i = 0; i < 18; ++i) {          // 48*48 = 576 float4 segments
    const int idx = tid + 32 * i;
    const int row = idx / 12, c4 = (idx % 12) * 4;
    async_b128(&sAttn[row * LDATT_ + c4], ab + row * DH_ + c4);
  }
  const float* vb = V + (size_t)bh * DH_ * N_;
#pragma unroll
  for (int i = 0; i < 24; ++i) {          // 48*64 = 768 float4 segments
    const int idx = tid + 32 * i;
    const int row = idx >> 4, c4 = (idx & 15) * 4;
    async_b128(&sV[row * 64 + c4], vb + row * N_ + n0 + c4);
  }
  WAIT_ASYNC_0();
  __syncthreads();

  const int lr = tid & 15;
  const int lh = tid >> 4;

  v8f c[3][4] = {};
#pragma unroll
  for (int k0 = 0; k0 < DH_; k0 += 4) {
    v2f bf[4];
#pragma unroll
    for (int nt = 0; nt < 4; ++nt) {
      bf[nt][0] = sV[(k0 + 2 * lh)     * 64 + nt * 16 + lr];
      bf[nt][1] = sV[(k0 + 2 * lh + 1) * 64 + nt * 16 + lr];
    }
#pragma unroll
    for (int dt = 0; dt < 3; ++dt) {
      v2f af = *(const v2f*)(&sAttn[(dt * 16 + lr) * LDATT_ + k0 + 2 * lh]);
#pragma unroll
      for (int nt = 0; nt < 4; ++nt) c[dt][nt] = wmma_f32(af, bf[nt], c[dt][nt]);
    }
  }

#pragma unroll
  for (int dt = 0; dt < 3; ++dt) {
#pragma unroll
    for (int nt = 0; nt < 4; ++nt) {
      const int n    = n0 + nt * 16 + lr;
      const int doff = h * DH_ + dt * 16 + 8 * lh;
      float* base = O + (size_t)(b * N_ + n) * C_ + doff;
      v4f lo = {c[dt][nt][0], c[dt][nt][1], c[dt][nt][2], c[dt][nt][3]};
      v4f hi = {c[dt][nt][4], c[dt][nt][5], c[dt][nt][6], c[dt][nt][7]};
      *(v4f*)(base)     = lo;
      *(v4f*)(base + 4) = hi;
    }
  }
}

// -------------------------------------------------------------------------
// Kernel 4: out = O @ w_proj^T + b_proj.  Same async double-buffered 128x128
// structure as kernel 1; natural-layout store with bias.
// -------------------------------------------------------------------------
__global__ __launch_bounds__(256) void k4_proj(const float* __restrict__ O,
                                               const float* __restrict__ W,
                                               const float* __restrict__ bias,
                                               float* __restrict__ Z) {
  __shared__ float sA[2][128 * LDA_];
  __shared__ float sB[2][128 * LDA_];

  const int tid  = threadIdx.x;
  const int lane = tid & 31;
  const int wave = tid >> 5;
  const int lr   = lane & 15;
  const int lh   = lane >> 4;

  const int mBase = blockIdx.y * 128;
  const int jBase = blockIdx.x * 128;

  const int r0 = tid >> 2;
  const int s0 = (tid & 3) * 4;
  const float* gA0 = O + (mBase + r0)      * C_ + s0;
  const float* gA1 = O + (mBase + r0 + 64) * C_ + s0;
  const float* gB0 = W + (jBase + r0)      * C_ + s0;
  const float* gB1 = W + (jBase + r0 + 64) * C_ + s0;

  v8f acc[8] = {};

  async_b128(&sA[0][r0 * LDA_ + s0],        gA0);
  async_b128(&sA[0][(r0 + 64) * LDA_ + s0], gA1);
  async_b128(&sB[0][r0 * LDA_ + s0],        gB0);
  async_b128(&sB[0][(r0 + 64) * LDA_ + s0], gB1);

  const int NK = C_ / KC_;
  for (int i = 0; i < NK; ++i) {
    if (i + 1 < NK) {
      const int kc = (i + 1) * KC_;
      const int nb = (i + 1) & 1;
      async_b128(&sA[nb][r0 * LDA_ + s0],        gA0 + kc);
      async_b128(&sA[nb][(r0 + 64) * LDA_ + s0], gA1 + kc);
      async_b128(&sB[nb][r0 * LDA_ + s0],        gB0 + kc);
      async_b128(&sB[nb][(r0 + 64) * LDA_ + s0], gB1 + kc);
      WAIT_ASYNC_4();
    } else {
      WAIT_ASYNC_0();
    }
    __syncthreads();

    const float* As = &sA[i & 1][(wave * 16 + lr) * LDA_ + 2 * lh];
    const float* Bs = &sB[i & 1][lr * LDA_ + 2 * lh];
#pragma unroll
    for (int kk = 0; kk < KC_; kk += 4) {
      v2f a = *(const v2f*)(As + kk);
#pragma unroll
      for (int nt = 0; nt < 8; ++nt) {
        v2f b = *(const v2f*)(Bs + nt * 16 * LDA_ + kk);
        acc[nt] = wmma_f32(a, b, acc[nt]);
      }
    }
    __syncthreads();
  }

  const int m0w = mBase + wave * 16;
#pragma unroll
  for (int nt = 0; nt < 8; ++nt) {
    const int j = jBase + nt * 16 + lr;
    const float bb = bias[j];
#pragma unroll
    for (int rr = 0; rr < 8; ++rr) {
      const int m = m0w + rr + 8 * lh;
      Z[(size_t)m * C_ + j] = acc[nt][rr] + bb;
    }
  }
}

// -------------------------------------------------------------------------
extern "C" void kernel_launch(void* const* d_in, const int* in_sizes, int n_in,
                              void* d_out, int out_size, void* d_ws, size_t ws_size,
                              hipStream_t stream) {
  const float* x     = (const float*)d_in[0];  // [B,N,C]
  const float* wqkv  = (const float*)d_in[1];  // [3C,C]
  const float* temp  = (const float*)d_in[2];  // [H,1,1]
  const float* wproj = (const float*)d_in[3];  // [C,C]
  const float* bproj = (const float*)d_in[4];  // [C]
  float* out = (float*)d_out;                  // [B,N,C]
  float* ws  = (float*)d_ws;

  const size_t SZ = (size_t)B_ * H_ * DH_ * N_;  // 38,535,168 floats
  float* q    = ws;
  float* k    = ws + SZ;
  float* v    = ws + 2 * SZ;
  float* attn = ws + 3 * SZ;                     // [B*H,48,48]
  float* O    = ws;                              // alias q (dead after k2)

  k1_qkv<<<dim3(C3_ / 128, M_ / 128), 256, 0, stream>>>(x, wqkv, q, k, v);
  k2_attn<<<dim3(B_ * H_), 288, 0, stream>>>(q, k, temp, attn);
  k3_av<<<dim3(N_ / 64, B_ * H_), 32, 0, stream>>>(attn, v, O);
  k4_proj<<<dim3(C_ / 128, M_ / 128), 256, 0, stream>>>(O, wproj, bproj, out);
}